// MLPDecoder_77592879169580
// MI455X (gfx1250) — compile-verified
//
#include <hip/hip_runtime.h>
#include <math.h>

typedef float v2f __attribute__((ext_vector_type(2)));
typedef float v8f __attribute__((ext_vector_type(8)));

// Problem dims (fixed by setup_inputs)
#define B_      4
#define NRAYS   4096
#define SPR     48
#define M_      (NRAYS*SPR)      // 196608 points per batch
#define NTOT    (B_*M_)          // 786432 total points
#define CH      32
#define PH      256
#define PW      256
#define HWU     (128*128)        // 16384
#define HID     64

// LDS layout (dword units, padded strides for 64-bank conflict freedom)
#define W1T_S   36
#define W2T_S   68
#define H_S     68
#define LDS_W1T 0
#define LDS_W2T (64*W1T_S)                 // 2304
#define LDS_H   (LDS_W2T + 16*W2T_S)       // 3392
#define LDS_TOT (LDS_H + 8*16*H_S)         // 12096 dwords = 48384 B

__global__ __launch_bounds__(256) void triplane_mlp_kernel(
    const float* __restrict__ points,
    const float* __restrict__ plane_axes,
    const float* __restrict__ plane_features,
    const float* __restrict__ uf_plane,
    const float* __restrict__ W1,
    const float* __restrict__ b1,
    const float* __restrict__ W2,
    const float* __restrict__ b2,
    const int*   __restrict__ subsample,
    float* __restrict__ out)
{
    __shared__ float lds[LDS_TOT];
    const int tid = threadIdx.x;

    // ---- cooperative staging: W1^T and zero-padded W2^T into LDS ----
    for (int i = tid; i < 64 * 32; i += 256) {
        int col = i >> 5, row = i & 31;                  // col: hidden, row: in-ch
        lds[LDS_W1T + col * W1T_S + row] = W1[row * 64 + col];
    }
    for (int i = tid; i < 16 * 64; i += 256) {
        int n = i >> 6, k = i & 63;                      // n: out col (pad 4->16), k: hidden
        lds[LDS_W2T + n * W2T_S + k] = (n < 4) ? W2[k * 4 + n] : 0.0f;
    }
    __syncthreads();

    const int lane   = tid & 31;
    const int m      = lane & 15;                        // WMMA row (this lane's point)
    const int koff   = (lane >> 4) << 1;                 // 0 or 2 (K pair within step)
    const int wlocal = tid >> 5;
    const long wave  = (long)blockIdx.x * 8 + wlocal;
    const long g     = wave * 16 + m;                    // global point id
    const int  b     = (int)(g / M_);
    const int  midx  = (int)(g - (long)b * M_);

    // ---- point & plane projections ----
    const float* pp = points + (size_t)g * 3;
    const float px = pp[0];                              // 2/BOX_WARP == 1.0
    const float py = pp[1];
    const float pz = pp[2];

    float acc[16];
    #pragma unroll
    for (int i = 0; i < 16; ++i) acc[i] = 0.0f;

    #pragma unroll
    for (int p = 0; p < 3; ++p) {
        const float* A = plane_axes + p * 9;
        float a00=A[0],a01=A[1],a02=A[2],a10=A[3],a11=A[4],a12=A[5],a20=A[6],a21=A[7],a22=A[8];
        float det = a00*(a11*a22-a12*a21) - a01*(a10*a22-a12*a20) + a02*(a10*a21-a11*a20);
        float id  = 1.0f / det;
        // need only columns 0,1 of inverse (adjugate/det)
        float i00 =  (a11*a22 - a12*a21) * id;
        float i10 = -(a10*a22 - a12*a20) * id;
        float i20 =  (a10*a21 - a11*a20) * id;
        float i01 = -(a01*a22 - a02*a21) * id;
        float i11 =  (a00*a22 - a02*a20) * id;
        float i21 = -(a00*a21 - a01*a20) * id;
        float gx = px*i00 + py*i10 + pz*i20;
        float gy = px*i01 + py*i11 + pz*i21;

        float u = (gx + 1.0f) * (PW * 0.5f) - 0.5f;
        float v = (gy + 1.0f) * (PH * 0.5f) - 0.5f;
        float x0f = floorf(u), y0f = floorf(v);
        float wx = u - x0f, wy = v - y0f;
        int ix0 = (int)x0f, iy0 = (int)y0f;

        const float* pf = plane_features + ((size_t)(b * 3 + p) * CH) * (size_t)(PH * PW);
        #pragma unroll
        for (int tap = 0; tap < 4; ++tap) {
            int xi = ix0 + (tap & 1);
            int yi = iy0 + (tap >> 1);
            float w = ((tap & 1) ? wx : (1.0f - wx)) * ((tap >> 1) ? wy : (1.0f - wy));
            bool valid = (xi >= 0) & (xi < PW) & (yi >= 0) & (yi < PH);
            w = valid ? w : 0.0f;
            int xc = min(max(xi, 0), PW - 1);
            int yc = min(max(yi, 0), PH - 1);
            const float* tp = pf + (size_t)yc * PW + xc;
            #pragma unroll
            for (int t = 0; t < 8; ++t) {
                int c0 = 4 * t + koff;                   // this lane's channel pair
                acc[2*t]     += w * tp[(size_t)c0       * (PH * PW)];
                acc[2*t + 1] += w * tp[(size_t)(c0 + 1) * (PH * PW)];
            }
        }
    }

    // ---- mean over planes + 0.01 * gathered upsample feature ----
    const int ray = midx / SPR;
    const int pix = subsample[b * NRAYS + ray];
    const float* ufp = uf_plane + (size_t)b * CH * HWU + pix;
    #pragma unroll
    for (int t = 0; t < 8; ++t) {
        int c0 = 4 * t + koff;
        acc[2*t]     = acc[2*t]     * (1.0f/3.0f) + 0.01f * ufp[(size_t)c0       * HWU];
        acc[2*t + 1] = acc[2*t + 1] * (1.0f/3.0f) + 0.01f * ufp[(size_t)(c0 + 1) * HWU];
    }

    // ---- layer 1: X(16x32) @ W1(32x64) via v_wmma_f32_16x16x4_f32 ----
    float* hreg = &lds[LDS_H + wlocal * (16 * H_S)];
    #pragma unroll
    for (int tn = 0; tn < 4; ++tn) {
        v8f c = {0.f,0.f,0.f,0.f,0.f,0.f,0.f,0.f};
        #pragma unroll
        for (int k = 0; k < 8; ++k) {
            v2f a; a.x = acc[2*k]; a.y = acc[2*k + 1];
            v2f bb = *(const v2f*)&lds[LDS_W1T + (16*tn + m) * W1T_S + 4*k + koff];
            c = __builtin_amdgcn_wmma_f32_16x16x4_f32(false, a, false, bb, (short)0, c, false, false);
        }
        float bias = b1[16*tn + m];                      // n == m == lane&15
        #pragma unroll
        for (int r = 0; r < 8; ++r) {
            float h = c[r] + bias;
            h = h > 0.0f ? h : 0.0f;                     // relu
            int mm = r + ((lane >> 4) << 3);             // D row for this vgpr
            hreg[mm * H_S + 16*tn + m] = h;
        }
    }
    // same-wave LDS ops are in-order; reads below see the stores above.

    // ---- layer 2: H(16x64) @ W2pad(64x16) via wmma, 16 K-steps ----
    v8f c2 = {0.f,0.f,0.f,0.f,0.f,0.f,0.f,0.f};
    #pragma unroll
    for (int k = 0; k < 16; ++k) {
        v2f a  = *(const v2f*)&hreg[m * H_S + 4*k + koff];
        v2f bb = *(const v2f*)&lds[LDS_W2T + m * W2T_S + 4*k + koff];
        c2 = __builtin_amdgcn_wmma_f32_16x16x4_f32(false, a, false, bb, (short)0, c2, false, false);
    }

    // ---- epilogue: sigma passthrough, rgb sigmoid scale ----
    const int n = m;                                     // output column for this lane
    if (n < 4) {
        float bias2 = b2[n];
        #pragma unroll
        for (int r = 0; r < 8; ++r) {
            int mm = r + ((lane >> 4) << 3);
            long gp = wave * 16 + mm;
            float v = c2[r] + bias2;
            if (n == 0) {
                out[gp] = v;                                         // sigma
            } else {
                float s = 1.0f / (1.0f + expf(-v));
                out[(long)NTOT + gp * 3 + (n - 1)] = s * 1.002f - 0.001f;  // rgb
            }
        }
    }
}

extern "C" void kernel_launch(void* const* d_in, const int* in_sizes, int n_in,
                              void* d_out, int out_size, void* d_ws, size_t ws_size,
                              hipStream_t stream) {
    const float* points         = (const float*)d_in[0];
    const float* plane_axes     = (const float*)d_in[1];
    const float* plane_features = (const float*)d_in[2];
    const float* uf_plane       = (const float*)d_in[3];
    const float* W1             = (const float*)d_in[4];
    const float* b1             = (const float*)d_in[5];
    const float* W2             = (const float*)d_in[6];
    const float* b2             = (const float*)d_in[7];
    const int*   subsample      = (const int*)d_in[8];
    float* out = (float*)d_out;

    dim3 grid(NTOT / (16 * 8));   // 16 points per wave, 8 waves per block -> 6144 blocks
    dim3 block(256);
    triplane_mlp_kernel<<<grid, block, 0, stream>>>(
        points, plane_axes, plane_features, uf_plane, W1, b1, W2, b2, subsample, out);
}